// GraphConvEncoder_20100446946052
// MI455X (gfx1250) — compile-verified
//
#include <hip/hip_runtime.h>
#include <hip/hip_bf16.h>

#define BN_EPS 1e-5f

typedef __attribute__((ext_vector_type(2))) float v2f;
typedef __attribute__((ext_vector_type(8))) float v8f;

// ---------------------------------------------------------------------------
// Scatter-add aggregation: agg[dst] += x[src], F features per node (F%4==0).
// One thread handles a float4 chunk of one edge.
// ---------------------------------------------------------------------------
__global__ void scatter_add_kernel(const float* __restrict__ x,
                                   const int* __restrict__ src,
                                   const int* __restrict__ dst,
                                   float* __restrict__ agg,
                                   int E, int F) {
    const int chunks = F >> 2;
    const long long total = (long long)E * chunks;
    for (long long i = (long long)blockIdx.x * blockDim.x + threadIdx.x;
         i < total;
         i += (long long)gridDim.x * blockDim.x) {
        const int e = (int)(i / chunks);
        const int c = (int)(i % chunks);
        const int s = src[e];
        const int d = dst[e];
        const float4 v = *(const float4*)(x + (size_t)s * F + (size_t)c * 4);
        float* p = agg + (size_t)d * F + (size_t)c * 4;
        atomicAdd(p + 0, v.x);
        atomicAdd(p + 1, v.y);
        atomicAdd(p + 2, v.z);
        atomicAdd(p + 3, v.w);
    }
}

// ---------------------------------------------------------------------------
// Dual GEMM via f32 WMMA: Out = A0 @ B0 + A1 @ B1 + bias
//   A0,A1 : [M, K] row-major     B0,B1 : [K, 128] row-major
//   Out   : [M, 128]
// One wave per 16x16 output tile. grid = (M/16, 128/16), block = 32.
//
// VGPR layouts per CDNA5 ISA 7.12.2 (16x16x4 f32):
//   A (16x4):  lane l, m = l&15, a[j] = A[m][k + 2*(l>>4) + j]
//   B (4x16):  lane l, n = l&15, b[j] = B[k + 2*(l>>4) + j][n]
//   C/D:       VGPR r, lane l -> row (r + 8*(l>>4)), col (l&15)
// ---------------------------------------------------------------------------
__global__ __launch_bounds__(32) void gemm_dual_wmma_kernel(
        const float* __restrict__ A0, const float* __restrict__ B0,
        const float* __restrict__ A1, const float* __restrict__ B1,
        const float* __restrict__ bias, float* __restrict__ Out,
        int M, int K) {
    const int lane = threadIdx.x;          // 0..31, wave32
    const int mt = blockIdx.x;
    const int nt = blockIdx.y;
    const int r16  = lane & 15;
    const int half = lane >> 4;            // 0 or 1
    const int kh   = half * 2;             // 0 or 2

    // Clamp A-row index for loads (EXEC must stay all-ones for WMMA).
    int m = mt * 16 + r16;
    if (m >= M) m = M - 1;
    const int n = nt * 16 + r16;           // 0..127, always in range

    v8f acc = {};

    const float* __restrict__ A = A0;
    const float* __restrict__ B = B0;
    #pragma unroll 1
    for (int pass = 0; pass < 2; ++pass) {
        #pragma unroll 4
        for (int k = 0; k < K; k += 4) {
            v2f a, b;
            a.x = A[(size_t)m * K + (k + kh + 0)];
            a.y = A[(size_t)m * K + (k + kh + 1)];
            b.x = B[(size_t)(k + kh + 0) * 128 + n];
            b.y = B[(size_t)(k + kh + 1) * 128 + n];
            // 8 args: (neg_a, A, neg_b, B, c_mod, C, reuse_a, reuse_b)
            acc = __builtin_amdgcn_wmma_f32_16x16x4_f32(
                false, a, false, b, (short)0, acc, false, false);
        }
        A = A1;
        B = B1;
    }

    const float bv = bias[n];
    const int mbase = mt * 16 + 8 * half;
    #pragma unroll
    for (int r = 0; r < 8; ++r) {
        const int row = mbase + r;
        if (row < M) {
            Out[(size_t)row * 128 + n] = acc[r] + bv;
        }
    }
}

// ---------------------------------------------------------------------------
// BatchNorm column statistics: stats[0:128] += sum, stats[128:256] += sumsq.
// block = 128 threads (one per column), each block strides over rows.
// ---------------------------------------------------------------------------
__global__ __launch_bounds__(128) void bn_stats_kernel(
        const float* __restrict__ h, float* __restrict__ stats, int M) {
    const int col = threadIdx.x;           // 0..127
    float s = 0.0f, s2 = 0.0f;
    for (int r = blockIdx.x; r < M; r += gridDim.x) {
        const float v = h[(size_t)r * 128 + col];
        s  += v;
        s2 += v * v;
    }
    atomicAdd(&stats[col], s);
    atomicAdd(&stats[128 + col], s2);
}

// ---------------------------------------------------------------------------
// BatchNorm apply (in place): h = (h - mean) * rsqrt(var + eps) * gamma + beta
// Optional ReLU. Population variance (E[x^2] - E[x]^2) matches jnp.var.
// ---------------------------------------------------------------------------
__global__ void bn_apply_kernel(float* __restrict__ h,
                                const float* __restrict__ stats,
                                const float* __restrict__ gamma,
                                const float* __restrict__ beta,
                                int M, int relu) {
    const long long total = (long long)M * 128;
    const float invM = 1.0f / (float)M;
    for (long long i = (long long)blockIdx.x * blockDim.x + threadIdx.x;
         i < total;
         i += (long long)gridDim.x * blockDim.x) {
        const int col = (int)(i & 127);
        const float mean = stats[col] * invM;
        const float var  = stats[128 + col] * invM - mean * mean;
        float v = (h[i] - mean) * rsqrtf(var + BN_EPS) * gamma[col] + beta[col];
        if (relu) v = fmaxf(v, 0.0f);
        h[i] = v;
    }
}

// ---------------------------------------------------------------------------
// Host-side launch. Inputs (in setup_inputs() order):
//  0 x[N,64] 1 edge_index[2,E] 2 W_rel0[64,128] 3 b_rel0[128] 4 W_root0[64,128]
//  5 gamma0  6 beta0  7 W_rel1[128,128] 8 b_rel1 9 W_root1[128,128]
// 10 gamma1 11 beta1                       Out: [N,128] f32
// ---------------------------------------------------------------------------
extern "C" void kernel_launch(void* const* d_in, const int* in_sizes, int n_in,
                              void* d_out, int out_size, void* d_ws, size_t ws_size,
                              hipStream_t stream) {
    const float* x      = (const float*)d_in[0];
    const int*   ei     = (const int*)d_in[1];
    const float* Wrel0  = (const float*)d_in[2];
    const float* brel0  = (const float*)d_in[3];
    const float* Wroot0 = (const float*)d_in[4];
    const float* gamma0 = (const float*)d_in[5];
    const float* beta0  = (const float*)d_in[6];
    const float* Wrel1  = (const float*)d_in[7];
    const float* brel1  = (const float*)d_in[8];
    const float* Wroot1 = (const float*)d_in[9];
    const float* gamma1 = (const float*)d_in[10];
    const float* beta1  = (const float*)d_in[11];
    float* out = (float*)d_out;

    const int M = in_sizes[0] / 64;     // 50000 nodes
    const int E = in_sizes[1] / 2;      // 800000 edges
    const int* src = ei;                // edge_index[0, :]
    const int* dst = ei + E;            // edge_index[1, :]

    // Workspace carve-up (floats): agg0[M*64] | h0[M*128] | agg1[M*128] | stats[256]
    float* agg0  = (float*)d_ws;
    float* h0    = agg0 + (size_t)M * 64;
    float* agg1  = h0   + (size_t)M * 128;
    float* stats = agg1 + (size_t)M * 128;

    hipMemsetAsync(agg0,  0, (size_t)M * 64  * sizeof(float), stream);
    hipMemsetAsync(agg1,  0, (size_t)M * 128 * sizeof(float), stream);
    hipMemsetAsync(stats, 0, 256 * sizeof(float), stream);

    const dim3 gemmGrid((M + 15) / 16, 8);
    const dim3 gemmBlock(32);

    // ---- Layer 0: conv -> BN -> ReLU ----
    {
        const long long work = (long long)E * 16;      // F=64 -> 16 float4 chunks
        const int blocks = (int)((work + 255) / 256);
        scatter_add_kernel<<<blocks, 256, 0, stream>>>(x, src, dst, agg0, E, 64);
    }
    gemm_dual_wmma_kernel<<<gemmGrid, gemmBlock, 0, stream>>>(
        agg0, Wrel0, x, Wroot0, brel0, h0, M, 64);
    bn_stats_kernel<<<512, 128, 0, stream>>>(h0, stats, M);
    {
        const long long total = (long long)M * 128;
        const int blocks = (int)((total + 255) / 256);
        bn_apply_kernel<<<blocks, 256, 0, stream>>>(h0, stats, gamma0, beta0, M, 1);
    }

    // ---- Layer 1: conv -> BN (no activation) ----
    {
        const long long work = (long long)E * 32;      // F=128 -> 32 float4 chunks
        const int blocks = (int)((work + 255) / 256);
        scatter_add_kernel<<<blocks, 256, 0, stream>>>(h0, src, dst, agg1, E, 128);
    }
    hipMemsetAsync(stats, 0, 256 * sizeof(float), stream);  // reuse stats buffer
    gemm_dual_wmma_kernel<<<gemmGrid, gemmBlock, 0, stream>>>(
        agg1, Wrel1, h0, Wroot1, brel1, out, M, 128);
    bn_stats_kernel<<<512, 128, 0, stream>>>(out, stats, M);
    {
        const long long total = (long long)M * 128;
        const int blocks = (int)((total + 255) / 256);
        bn_apply_kernel<<<blocks, 256, 0, stream>>>(out, stats, gamma1, beta1, M, 0);
    }
}